// FDSyn_GNN_84602265796939
// MI455X (gfx1250) — compile-verified
//
#include <hip/hip_runtime.h>
#include <math.h>

typedef float v2f __attribute__((ext_vector_type(2)));
typedef float v8f __attribute__((ext_vector_type(8)));

#define B_  32
#define N_  400
#define T_  200
#define H_  128
#define HD_ 64
#define L_  4

// ---------------------------------------------------------------------------
// Generic f32 WMMA GEMM:  C = alpha * A x op(W) + bias
//   transB=1 : C[m,n] = sum_k A[m,k] * W[n,k]   (W is [Nout,K], row-major)
//   transB=0 : C[m,n] = sum_k A[m,k] * W[k,n]   (W is [K,Nout], row-major)
// REQUIREMENTS (all call sites verified): M%16==0, Nout%16==0, K%4==0,
// lda/ldw even on contiguous paths (8B-aligned v2f fragment loads).
// 4 waves / block, one 16x16 C tile per wave, flat tile index with
// wave-uniform early exit (EXEC all-ones through every WMMA, per ISA 7.12).
// Fragment layouts per CDNA5 ISA 7.12.2:
//   A(16x4): lanes 0-15 -> M=lane, VGPR0=K0,VGPR1=K1 ; lanes 16-31 -> K2,K3
//   B(4x16): VGPR v, lanes 0-15 -> K=v,N=lane ; lanes 16-31 -> K=v+2,N=lane-16
//   C/D(16x16): VGPR r, lanes 0-15 -> M=r ; lanes 16-31 -> M=r+8 ; N=lane%16
// ---------------------------------------------------------------------------
__global__ __launch_bounds__(128)
void wmma_gemm_f32(const float* __restrict__ A, const float* __restrict__ W,
                   const float* __restrict__ bias, float* __restrict__ C,
                   int M, int K, int Nout, int lda, int ldw, int ldc,
                   int nH,
                   long long sAb, long long sAh,
                   long long sWb, long long sWh,
                   long long sCb, long long sCh, long long sBias,
                   float alpha, int transB)
{
  int z  = blockIdx.z;
  int zb = z / nH, zh = z - zb * nH;
  A += (long long)zb * sAb + (long long)zh * sAh;
  W += (long long)zb * sWb + (long long)zh * sWh;
  C += (long long)zb * sCb + (long long)zh * sCh;

  int lane = threadIdx.x & 31;
  int wid  = threadIdx.x >> 5;
  int ntn    = Nout >> 4;
  int nTiles = (M >> 4) * ntn;
  int tile   = blockIdx.x * 4 + wid;
  if (tile >= nTiles) return;          // wave-uniform exit
  int tm = tile / ntn, tn = tile - tm * ntn;

  int l16  = lane & 15;
  int half = lane >> 4;

  const float* Ap = A + (long long)(tm * 16 + l16) * lda + 2 * half;
  v8f acc = {};

  if (transB) {
    const float* Wp = W + (long long)(tn * 16 + l16) * ldw + 2 * half;
#pragma unroll 4
    for (int kk = 0; kk < K; kk += 4) {
      v2f af = *(const v2f*)(Ap + kk);
      v2f bf = *(const v2f*)(Wp + kk);
      acc = __builtin_amdgcn_wmma_f32_16x16x4_f32(false, af, false, bf,
                                                  (short)0, acc, false, false);
    }
  } else {
    const float* Wp = W + (long long)(2 * half) * ldw + (tn * 16 + l16);
#pragma unroll 4
    for (int kk = 0; kk < K; kk += 4) {
      v2f af = *(const v2f*)(Ap + kk);
      v2f bf;
      bf.x = Wp[(long long)kk * ldw];
      bf.y = Wp[(long long)(kk + 1) * ldw];
      acc = __builtin_amdgcn_wmma_f32_16x16x4_f32(false, af, false, bf,
                                                  (short)0, acc, false, false);
    }
  }

  int col  = tn * 16 + l16;
  float bv = bias ? (bias + (long long)z * sBias)[col] : 0.f;
  float* Cp = C + (long long)(tm * 16 + 8 * half) * ldc + col;
#pragma unroll
  for (int r = 0; r < 8; ++r)
    Cp[(long long)r * ldc] = alpha * acc[r] + bv;
}

// ---------------------------------------------------------------------------
// Bidirectional GRU scan. One block per (batch, direction), 128 threads.
// xw: [2][B*T][384] precomputed input gates (r,z,n).  go: [B*T][256].
// ---------------------------------------------------------------------------
__global__ __launch_bounds__(128)
void gru_scan_kernel(const float* __restrict__ xw, const float* __restrict__ whh,
                     const float* __restrict__ bhh, float* __restrict__ go)
{
  int b = blockIdx.x;
  int d = blockIdx.y;
  int j = threadIdx.x;
  __shared__ float hcur[H_];
  hcur[j] = 0.f;
  __syncthreads();

  const float* Wd  = whh + (long long)d * 3 * H_ * H_;
  const float* bd  = bhh + d * 3 * H_;
  const float* xwd = xw + ((long long)d * B_ * T_ + (long long)b * T_) * (3 * H_);

  for (int s = 0; s < T_; ++s) {
    int t = d ? (T_ - 1 - s) : s;
    const float* xt = xwd + (long long)t * (3 * H_);
    float hr = bd[j], hz = bd[H_ + j], hn = bd[2 * H_ + j];
    const float* wr = Wd + (long long)j * H_;
    const float* wz = Wd + (long long)(H_ + j) * H_;
    const float* wn = Wd + (long long)(2 * H_ + j) * H_;
#pragma unroll 4
    for (int k2 = 0; k2 < H_; ++k2) {
      float hk = hcur[k2];
      hr += wr[k2] * hk;
      hz += wz[k2] * hk;
      hn += wn[k2] * hk;
    }
    float r  = 1.f / (1.f + expf(-(xt[j] + hr)));
    float zt = 1.f / (1.f + expf(-(xt[H_ + j] + hz)));
    float nn = tanhf(xt[2 * H_ + j] + r * hn);
    float h2 = (1.f - zt) * nn + zt * hcur[j];
    __syncthreads();
    hcur[j] = h2;
    __syncthreads();
    go[((long long)b * T_ + t) * (2 * H_) + d * H_ + j] = h2;
  }
}

// ---------------------------------------------------------------------------
// enc: mean & max over T, then min-max normalize each to [-1,1] over N per b.
// ---------------------------------------------------------------------------
__global__ __launch_bounds__(512)
void enc_kernel(const float* __restrict__ gproj,
                float* __restrict__ enc0, float* __restrict__ enc1)
{
  int b = blockIdx.x, tid = threadIdx.x;
  float meanv = 0.f, maxv = -INFINITY;
  if (tid < N_) {
    const float* col = gproj + (long long)b * T_ * N_ + tid;
    for (int t = 0; t < T_; ++t) {
      float x = col[(long long)t * N_];
      meanv += x;
      maxv = fmaxf(maxv, x);
    }
    meanv *= (1.f / (float)T_);
  }
  __shared__ float red[512];
  float mn0, mx0, mn1, mx1;

  red[tid] = (tid < N_) ? meanv : INFINITY; __syncthreads();
  for (int s = 256; s > 0; s >>= 1) { if (tid < s) red[tid] = fminf(red[tid], red[tid + s]); __syncthreads(); }
  mn0 = red[0]; __syncthreads();
  red[tid] = (tid < N_) ? meanv : -INFINITY; __syncthreads();
  for (int s = 256; s > 0; s >>= 1) { if (tid < s) red[tid] = fmaxf(red[tid], red[tid + s]); __syncthreads(); }
  mx0 = red[0]; __syncthreads();
  red[tid] = (tid < N_) ? maxv : INFINITY; __syncthreads();
  for (int s = 256; s > 0; s >>= 1) { if (tid < s) red[tid] = fminf(red[tid], red[tid + s]); __syncthreads(); }
  mn1 = red[0]; __syncthreads();
  red[tid] = (tid < N_) ? maxv : -INFINITY; __syncthreads();
  for (int s = 256; s > 0; s >>= 1) { if (tid < s) red[tid] = fmaxf(red[tid], red[tid + s]); __syncthreads(); }
  mx1 = red[0]; __syncthreads();

  if (tid < N_) {
    enc0[b * N_ + tid] = 2.f * (meanv - mn0) / (mx0 - mn0 + 1e-8f) - 1.f;
    enc1[b * N_ + tid] = 2.f * (maxv  - mn1) / (mx1 - mn1 + 1e-8f) - 1.f;
  }
}

// add the two enc columns of initW (K tail 400..401) into h
__global__ void init2_kernel(float* __restrict__ h, const float* __restrict__ enc0,
                             const float* __restrict__ enc1, const float* __restrict__ initW,
                             long long n)
{
  for (long long i = (long long)blockIdx.x * blockDim.x + threadIdx.x; i < n;
       i += (long long)gridDim.x * blockDim.x) {
    int j = (int)(i & (H_ - 1));
    long long row = i >> 7;
    h[i] += enc0[row] * initW[j * (N_ + 2) + N_] + enc1[row] * initW[j * (N_ + 2) + N_ + 1];
  }
}

// ---------------------------------------------------------------------------
// Exact 70th percentile per sample (jnp.percentile linear interpolation):
// bisection on order-preserving uint mapping to find k-th & (k+1)-th smallest.
// ---------------------------------------------------------------------------
__global__ __launch_bounds__(256)
void percentile_kernel(const float* __restrict__ a, float* __restrict__ thr)
{
  const int M = N_ * N_;
  int b = blockIdx.x;
  const float* ab = a + (long long)b * M;
  __shared__ unsigned scnt[256];
  __shared__ unsigned s_lo, s_hi;
  __shared__ float vks[2];

  double pos  = 0.70 * (double)(M - 1);
  long long k0 = (long long)pos;         // 111999
  double frac = pos - (double)k0;        // 0.3

  for (int which = 0; which < 2; ++which) {
    unsigned long long krank = (unsigned long long)(k0 + which);
    if (threadIdx.x == 0) { s_lo = 0u; s_hi = 0xFFFFFFFFu; }
    __syncthreads();
    for (int it = 0; it < 32; ++it) {
      unsigned lo = s_lo, hi = s_hi;
      if (lo >= hi) break;          // uniform across block
      unsigned mid = lo + ((hi - lo) >> 1);
      unsigned cnt = 0;
      for (int i = threadIdx.x; i < M; i += 256) {
        unsigned u = __float_as_uint(ab[i]);
        u = (u & 0x80000000u) ? ~u : (u | 0x80000000u);
        cnt += (u <= mid) ? 1u : 0u;
      }
      scnt[threadIdx.x] = cnt; __syncthreads();
      for (int s = 128; s > 0; s >>= 1) { if (threadIdx.x < s) scnt[threadIdx.x] += scnt[threadIdx.x + s]; __syncthreads(); }
      if (threadIdx.x == 0) {
        if ((unsigned long long)scnt[0] >= krank + 1ull) s_hi = mid; else s_lo = mid + 1u;
      }
      __syncthreads();
    }
    if (threadIdx.x == 0) {
      unsigned u = s_lo;
      unsigned bits = (u & 0x80000000u) ? (u ^ 0x80000000u) : ~u;
      vks[which] = __uint_as_float(bits);
    }
    __syncthreads();
  }
  if (threadIdx.x == 0)
    thr[b] = (float)((1.0 - frac) * (double)vks[0] + frac * (double)vks[1]);
}

// masked softmax over last dim of scores[b][h][i][:]
__global__ __launch_bounds__(256)
void softmax_mask_kernel(float* __restrict__ scores, const float* __restrict__ a,
                         const float* __restrict__ thr)
{
  int i = blockIdx.x, hh = blockIdx.y, b = blockIdx.z;
  float* row = scores + (((long long)b * 2 + hh) * N_ + i) * N_;
  const float* arow = a + ((long long)b * N_ + i) * N_;
  float tb = thr[b];
  __shared__ float red[256];
  int tid = threadIdx.x;

  float mx = -INFINITY;
  for (int j = tid; j < N_; j += 256) {
    float s = (arow[j] > tb) ? row[j] : -INFINITY;
    row[j] = s;
    mx = fmaxf(mx, s);
  }
  red[tid] = mx; __syncthreads();
  for (int s = 128; s > 0; s >>= 1) { if (tid < s) red[tid] = fmaxf(red[tid], red[tid + s]); __syncthreads(); }
  mx = red[0]; __syncthreads();

  float sum = 0.f;
  for (int j = tid; j < N_; j += 256) {
    float e2 = expf(row[j] - mx);
    row[j] = e2; sum += e2;
  }
  red[tid] = sum; __syncthreads();
  for (int s = 128; s > 0; s >>= 1) { if (tid < s) red[tid] += red[tid + s]; __syncthreads(); }
  float inv = 1.f / red[0];
  for (int j = tid; j < N_; j += 256) row[j] *= inv;
}

__global__ void eps_add_kernel(float* __restrict__ y, const float* __restrict__ h,
                               const float* __restrict__ eps, int l, long long n)
{
  float e = eps[l];
  for (long long i = (long long)blockIdx.x * blockDim.x + threadIdx.x; i < n;
       i += (long long)gridDim.x * blockDim.x)
    y[i] += e * h[i];
}

// dst = act(bn(src)); act=0 relu, act=1 exact gelu. channel = i % 128, param row l.
__global__ void bn_act_kernel(float* __restrict__ dst, const float* __restrict__ src,
                              const float* __restrict__ g, const float* __restrict__ bb,
                              const float* __restrict__ mm, const float* __restrict__ vv,
                              int l, int act, long long n)
{
  for (long long i = (long long)blockIdx.x * blockDim.x + threadIdx.x; i < n;
       i += (long long)gridDim.x * blockDim.x) {
    int j = (int)(i & (H_ - 1));
    int p = l * H_ + j;
    float y = (src[i] - mm[p]) * rsqrtf(vv[p] + 1e-5f) * g[p] + bb[p];
    if (act == 0) y = fmaxf(y, 0.f);
    else          y = 0.5f * y * (1.f + erff(y * 0.70710678118654752440f));
    dst[i] = y;
  }
}

__global__ void mean_nodes_kernel(const float* __restrict__ h, float* __restrict__ meanh)
{
  int b = blockIdx.x, j = threadIdx.x;
  const float* p = h + (long long)b * N_ * H_ + j;
  float s = 0.f;
  for (int n2 = 0; n2 < N_; ++n2) s += p[(long long)n2 * H_];
  meanh[b * H_ + j] = s * (1.f / (float)N_);
}

__global__ void sigmoid_dual_kernel(float* __restrict__ g, float* __restrict__ outA,
                                    int l, int n)
{
  int i = blockIdx.x * blockDim.x + threadIdx.x;
  if (i < n) {
    int b = i / N_, nn = i - b * N_;
    float v = 1.f / (1.f + expf(-g[i]));
    g[i] = v;
    outA[((long long)b * L_ + l) * N_ + nn] = v;
  }
}

__global__ void readout_kernel(const float* __restrict__ h, const float* __restrict__ g,
                               float* __restrict__ rd, float* __restrict__ outL, int l)
{
  int b = blockIdx.x, j = threadIdx.x;
  const float* hp = h + (long long)b * N_ * H_ + j;
  const float* gp = g + b * N_;
  float s = 0.f;
  for (int n2 = 0; n2 < N_; ++n2) s += hp[(long long)n2 * H_] * gp[n2];
  float v = s * (1.f / (float)N_);
  rd[b * H_ + j] = v;
  outL[((long long)b * L_ + l) * H_ + j] = v;
}

__global__ void logit_kernel(const float* __restrict__ rd, const float* __restrict__ hW,
                             const float* __restrict__ hb2, float* __restrict__ out, int l)
{
  int tid = threadIdx.x;
  if (tid >= 2 * B_) return;
  int b = tid >> 1, c = tid & 1;
  const float* w = hW + ((long long)l * 2 + c) * H_;
  const float* r = rd + b * H_;
  float s = hb2[l * 2 + c];
  for (int k2 = 0; k2 < H_; ++k2) s += r[k2] * w[k2];
  out[b * 2 + c] += s;
}

__global__ void zero_kernel(float* __restrict__ p, int n)
{
  int i = blockIdx.x * blockDim.x + threadIdx.x;
  if (i < n) p[i] = 0.f;
}

// ---------------------------------------------------------------------------
static void gemm(hipStream_t s, const float* A, const float* W, const float* bias, float* C,
                 int M, int K, int N, int lda, int ldw, int ldc,
                 int batch = 1, int nH = 1,
                 long long sAb = 0, long long sAh = 0,
                 long long sWb = 0, long long sWh = 0,
                 long long sCb = 0, long long sCh = 0, long long sBias = 0,
                 float alpha = 1.f, int transB = 1)
{
  int nTiles = (M / 16) * (N / 16);
  dim3 g((unsigned)((nTiles + 3) / 4), 1u, (unsigned)batch);
  wmma_gemm_f32<<<g, dim3(128), 0, s>>>(A, W, bias, C, M, K, N, lda, ldw, ldc, nH,
                                        sAb, sAh, sWb, sWh, sCb, sCh, sBias, alpha, transB);
}

extern "C" void kernel_launch(void* const* d_in, const int* in_sizes, int n_in,
                              void* d_out, int out_size, void* d_ws, size_t ws_size,
                              hipStream_t stream)
{
  const float* a     = (const float*)d_in[1];
  const float* t     = (const float*)d_in[2];
  const float* wih   = (const float*)d_in[3];
  const float* whh   = (const float*)d_in[4];
  const float* bih   = (const float*)d_in[5];
  const float* bhh   = (const float*)d_in[6];
  const float* projW = (const float*)d_in[7];
  const float* projb = (const float*)d_in[8];
  const float* initW = (const float*)d_in[9];
  const float* initb = (const float*)d_in[10];
  const float* qW  = (const float*)d_in[11];
  const float* kW  = (const float*)d_in[12];
  const float* vW  = (const float*)d_in[13];
  const float* oW  = (const float*)d_in[14];
  const float* m1W = (const float*)d_in[15];
  const float* m2W = (const float*)d_in[16];
  const float* seW = (const float*)d_in[17];
  const float* qb  = (const float*)d_in[18];
  const float* kb  = (const float*)d_in[19];
  const float* vb  = (const float*)d_in[20];
  const float* ob  = (const float*)d_in[21];
  const float* m1b = (const float*)d_in[22];
  const float* m2b = (const float*)d_in[23];
  const float* seb = (const float*)d_in[24];
  const float* eps = (const float*)d_in[25];
  const float* bn1g = (const float*)d_in[26];
  const float* bn2g = (const float*)d_in[27];
  const float* sbg  = (const float*)d_in[28];
  const float* bn1b = (const float*)d_in[29];
  const float* bn2b = (const float*)d_in[30];
  const float* sbb  = (const float*)d_in[31];
  const float* bn1m = (const float*)d_in[32];
  const float* bn2m = (const float*)d_in[33];
  const float* sbm  = (const float*)d_in[34];
  const float* bn1v = (const float*)d_in[35];
  const float* bn2v = (const float*)d_in[36];
  const float* sbv  = (const float*)d_in[37];
  const float* saW  = (const float*)d_in[38];
  const float* sab  = (const float*)d_in[39];
  const float* hW   = (const float*)d_in[40];
  const float* hb   = (const float*)d_in[41];

  float* out       = (float*)d_out;
  float* out_logit = out;                         // [B,2]
  float* out_attn  = out + 2 * B_;                // [B,L,N]
  float* out_lat   = out + 2 * B_ + B_ * L_ * N_; // [B,L,H]

  // workspace layout (floats), with manual reuse (~97 MB total)
  float* ws = (float*)d_ws;
  long long off = 0;
  float* xw     = ws + off; off += 2LL * B_ * T_ * 3 * H_;        // reused: q,k,v
  float* go     = ws + off; off += (long long)B_ * T_ * 2 * H_;
  float* gproj  = ws + off; off += (long long)B_ * T_ * N_;       // reused: h2
  float* hbuf   = ws + off; off += (long long)B_ * N_ * H_;
  float* scores = ws + off; off += (long long)B_ * 2 * N_ * N_;
  float* vagg   = ws + off; off += (long long)B_ * N_ * H_;
  float* mbuf   = ws + off; off += (long long)B_ * N_ * H_;
  float* enc0   = ws + off; off += (long long)B_ * N_;
  float* enc1   = ws + off; off += (long long)B_ * N_;
  float* thr    = ws + off; off += B_;
  float* meanh  = ws + off; off += B_ * H_;
  float* ebuf   = ws + off; off += B_ * H_;
  float* gtmp   = ws + off; off += B_ * N_;
  float* rdbuf  = ws + off; off += B_ * H_;
  float* qbuf = xw;
  float* kbuf = xw + (long long)B_ * N_ * H_;
  float* vbuf = xw + 2LL * B_ * N_ * H_;
  float* h2   = gproj;   // alias; gproj dead after enc_kernel

  const long long nBNH = (long long)B_ * N_ * H_;   // 1,638,400
  const int MT = B_ * T_;   // 6400
  const int MN = B_ * N_;   // 12800

  // ---- GRU input GEMM: xw[d] = t @ wih[d]^T + bih[d]   (batch over dir) ----
  gemm(stream, t, wih, bih, xw, MT, N_, 3 * H_, N_, N_, 3 * H_,
       /*batch*/2, /*nH*/1, 0, 0, (long long)3 * H_ * N_, 0,
       (long long)MT * 3 * H_, 0, 3 * H_, 1.f, 1);

  // ---- GRU scan (both dirs) -> go [B*T, 256] ----
  gru_scan_kernel<<<dim3(B_, 2), dim3(H_), 0, stream>>>(xw, whh, bhh, go);

  // ---- proj: gproj = go @ projW^T + projb ----
  gemm(stream, go, projW, projb, gproj, MT, 2 * H_, N_, 2 * H_, 2 * H_, N_);

  // ---- enc (mean/max over T, minmax over N) ----
  enc_kernel<<<dim3(B_), dim3(512), 0, stream>>>(gproj, enc0, enc1);

  // ---- h = [a | enc] @ initW^T + initb  (split: a-part GEMM + enc tail) ----
  gemm(stream, a, initW, initb, hbuf, MN, N_, H_, N_, N_ + 2, H_);
  init2_kernel<<<dim3(4096), dim3(256), 0, stream>>>(hbuf, enc0, enc1, initW, nBNH);

  // ---- per-sample percentile threshold ----
  percentile_kernel<<<dim3(B_), dim3(256), 0, stream>>>(a, thr);

  // ---- zero logits ----
  zero_kernel<<<dim3(1), dim3(64), 0, stream>>>(out_logit, 2 * B_);

  const int ewBlocks = (int)((nBNH + 255) / 256);

  for (int l = 0; l < L_; ++l) {
    const long long wOff = (long long)l * H_ * H_;
    const long long bOff = (long long)l * H_;

    // q, k, v projections
    gemm(stream, hbuf, qW + wOff, qb + bOff, qbuf, MN, H_, H_, H_, H_, H_);
    gemm(stream, hbuf, kW + wOff, kb + bOff, kbuf, MN, H_, H_, H_, H_, H_);
    gemm(stream, hbuf, vW + wOff, vb + bOff, vbuf, MN, H_, H_, H_, H_, H_);

    // scores[b,h] = (q[b,:,h] @ k[b,:,h]^T) / 8   (batch = B*HEADS)
    gemm(stream, qbuf, kbuf, nullptr, scores, N_, HD_, N_, H_, H_, N_,
         /*batch*/B_ * 2, /*nH*/2,
         (long long)N_ * H_, HD_, (long long)N_ * H_, HD_,
         (long long)2 * N_ * N_, (long long)N_ * N_, 0,
         0.125f, 1);

    // mask + softmax (in place)
    softmax_mask_kernel<<<dim3(N_, 2, B_), dim3(256), 0, stream>>>(scores, a, thr);

    // vagg[b,:,h] = attn[b,h] @ v[b,:,h]   (transB=0)
    gemm(stream, scores, vbuf, nullptr, vagg, N_, N_, HD_, N_, H_, H_,
         /*batch*/B_ * 2, /*nH*/2,
         (long long)2 * N_ * N_, (long long)N_ * N_,
         (long long)N_ * H_, HD_,
         (long long)N_ * H_, HD_, 0,
         1.f, 0);

    // output projection + eps*h
    gemm(stream, vagg, oW + wOff, ob + bOff, h2, MN, H_, H_, H_, H_, H_);
    eps_add_kernel<<<dim3(ewBlocks), dim3(256), 0, stream>>>(h2, hbuf, eps, l, nBNH);

    // MLP: m = relu(bn1(h2 @ m1W^T + m1b)) ; h = relu(bn2(m @ m2W^T + m2b))
    gemm(stream, h2, m1W + wOff, m1b + bOff, mbuf, MN, H_, H_, H_, H_, H_);
    bn_act_kernel<<<dim3(ewBlocks), dim3(256), 0, stream>>>(mbuf, mbuf, bn1g, bn1b, bn1m, bn1v, l, 0, nBNH);
    gemm(stream, mbuf, m2W + wOff, m2b + bOff, h2, MN, H_, H_, H_, H_, H_);
    bn_act_kernel<<<dim3(ewBlocks), dim3(256), 0, stream>>>(hbuf, h2, bn2g, bn2b, bn2m, bn2v, l, 0, nBNH);

    // SERO readout
    mean_nodes_kernel<<<dim3(B_), dim3(H_), 0, stream>>>(hbuf, meanh);
    gemm(stream, meanh, seW + wOff, seb + bOff, ebuf, B_, H_, H_, H_, H_, H_);
    bn_act_kernel<<<dim3(16), dim3(256), 0, stream>>>(ebuf, ebuf, sbg, sbb, sbm, sbv, l, 1, (long long)B_ * H_);
    gemm(stream, ebuf, saW + (long long)l * N_ * H_, sab + (long long)l * N_, gtmp,
         B_, H_, N_, H_, H_, N_);
    sigmoid_dual_kernel<<<dim3((B_ * N_ + 255) / 256), dim3(256), 0, stream>>>(gtmp, out_attn, l, B_ * N_);
    readout_kernel<<<dim3(B_), dim3(H_), 0, stream>>>(hbuf, gtmp, rdbuf, out_lat, l);
    logit_kernel<<<dim3(1), dim3(64), 0, stream>>>(rdbuf, hW, hb, out_logit, l);
  }
}